// EnhancedSyntheticEmbeddingLayer_37984690766005
// MI455X (gfx1250) — compile-verified
//
#include <hip/hip_runtime.h>
#include <math.h>

// Problem constants (from reference setup_inputs)
#define BB 8
#define SS 1024
#define DDIM 512
#define NSTATE 16
#define NLAYERS 3
#define NHEADS 4
#define HDIM 128          // DDIM / NHEADS
#define BSD (BB*SS*DDIM)  // 4,194,304
#define BSN (BB*SS*NSTATE)

typedef __attribute__((ext_vector_type(2))) float v2f;
typedef __attribute__((ext_vector_type(8))) float v8f;

// Explicit global-address-space types: loop-carried pointer PHIs defeat
// clang's address-space inference (they lowered to flat_load_b32 + xcnt
// bookkeeping), so we force addrspace(1) at kernel entry.
typedef __attribute__((address_space(1))) float gfloat;
typedef __attribute__((address_space(1))) v2f   gv2f;

// ---------------------------------------------------------------------------
// WMMA fp32 GEMM:  C[M,N] = A[M,K] @ B[K,N] (+ bias[n])
// Row-major.  Each wave owns an (MT*16 x NT*16) output block built from
// MT*NT accumulators, stepping K by 4 via V_WMMA_F32_16X16X4_F32.
// Fragment layouts per CDNA5 ISA 7.12.2:
//   A 16x4 : lane L -> m=L&15, holds K = 2*(L>>4)+{0,1}   (8B contiguous)
//   B 4x16 : lane L -> n=L&15, holds K = 2*(L>>4)+{0,1}
//   C/D    : vgpr r -> m = r + 8*(L>>4), n = L&15
// B pointers advance by 4*ldb per step (strength-reduced addressing).
// ---------------------------------------------------------------------------
template<int MT, int NT>
__global__ __launch_bounds__(256)
void gemm_wmma(const float* __restrict__ Ap, const float* __restrict__ Bp,
               const float* __restrict__ biasp, float* __restrict__ Cp,
               int M, int K, int N, int lda, int ldb, int ldc)
{
    const gfloat* A = (const gfloat*)Ap;
    const gfloat* B = (const gfloat*)Bp;
    gfloat*       C = (gfloat*)Cp;

    const int lane = threadIdx.x & 31;
    const int wave = blockIdx.x * (blockDim.x >> 5) + (threadIdx.x >> 5);
    const int ntn  = N / (16 * NT);
    const int tile_m = wave / ntn;
    const int tile_n = wave - tile_m * ntn;
    if (tile_m >= M / (16 * MT)) return;     // wave-uniform branch, EXEC stays full

    const int m0 = tile_m * 16 * MT + (lane & 15);
    const int n0 = tile_n * 16 * NT + (lane & 15);
    const int kb = (lane >> 4) * 2;          // 0 or 2

    const gfloat* pA[MT];
#pragma unroll
    for (int i = 0; i < MT; ++i)
        pA[i] = A + (size_t)(m0 + 16 * i) * lda + kb;

    const gfloat* pB[NT];
#pragma unroll
    for (int j = 0; j < NT; ++j)
        pB[j] = B + (size_t)kb * ldb + n0 + 16 * j;

    const size_t ldb4 = 4 * (size_t)ldb;

    v8f acc[MT][NT];
    const v8f vzero = {0.f,0.f,0.f,0.f,0.f,0.f,0.f,0.f};
#pragma unroll
    for (int i = 0; i < MT; ++i)
#pragma unroll
        for (int j = 0; j < NT; ++j) acc[i][j] = vzero;

#pragma unroll 4
    for (int k0 = 0; k0 < K; k0 += 4) {
        v2f a[MT];
#pragma unroll
        for (int i = 0; i < MT; ++i)
            a[i] = *(const gv2f*)(pA[i] + k0);

        v2f b[NT];
#pragma unroll
        for (int j = 0; j < NT; ++j) {
            b[j].x = pB[j][0];
            b[j].y = pB[j][ldb];
            pB[j] += ldb4;
        }

#pragma unroll
        for (int i = 0; i < MT; ++i)
#pragma unroll
            for (int j = 0; j < NT; ++j)
                acc[i][j] = __builtin_amdgcn_wmma_f32_16x16x4_f32(
                                false, a[i], false, b[j], (short)0,
                                acc[i][j], false, false);
    }

    const int rbase = (lane >> 4) * 8;
#pragma unroll
    for (int j = 0; j < NT; ++j) {
        const int n = n0 + 16 * j;
        const float bn = biasp ? ((const gfloat*)biasp)[n] : 0.0f;
#pragma unroll
        for (int i = 0; i < MT; ++i) {
            const int mb = tile_m * 16 * MT + 16 * i + rbase;
#pragma unroll
            for (int r = 0; r < 8; ++r)
                C[(size_t)(mb + r) * ldc + n] = acc[i][j][r] + bn;
        }
    }
}

// ---------------------------------------------------------------------------
// Elementwise kernels
// ---------------------------------------------------------------------------
__global__ void k_mask_mul(const float* __restrict__ x,
                           const unsigned char* __restrict__ mask,
                           float* __restrict__ y, int total)
{
    int i = blockIdx.x * blockDim.x + threadIdx.x;
    if (i < total) y[i] = mask[i / DDIM] ? x[i] : 0.0f;
}

__global__ void k_softplus(float* __restrict__ v, int total)
{
    int i = blockIdx.x * blockDim.x + threadIdx.x;
    if (i < total) {
        float x = v[i];
        v[i] = (x > 20.0f) ? x : log1pf(expf(x));
    }
}

__global__ void k_fill(float* __restrict__ v, float val, int total)
{
    int i = blockIdx.x * blockDim.x + threadIdx.x;
    if (i < total) v[i] = val;
}

__global__ void k_scale(float* __restrict__ v, float sc, int total)
{
    int i = blockIdx.x * blockDim.x + threadIdx.x;
    if (i < total) v[i] *= sc;
}

__global__ void k_liquid_acc(float* __restrict__ acc,
                             const float* __restrict__ hbn,
                             const float* __restrict__ lin,
                             const float* __restrict__ tau, int total)
{
    int i = blockIdx.x * blockDim.x + threadIdx.x;
    if (i < total) {
        int d = i % DDIM;
        float g = 1.0f / (1.0f + expf(-tau[d]));
        acc[i] += hbn[i] * g + tanhf(lin[i]) * (1.0f - g);
    }
}

__global__ void k_metab(float* __restrict__ h, const float* __restrict__ G,
                        const float* __restrict__ Mm, int total)
{
    int i = blockIdx.x * blockDim.x + threadIdx.x;
    if (i < total)
        h[i] = h[i] + (1.0f / (1.0f + expf(-G[i]))) * Mm[i];
}

__global__ void k_ternary(const float* __restrict__ W,
                          const float* __restrict__ sp,
                          float* __restrict__ Wq, int n)
{
    int i = blockIdx.x * blockDim.x + threadIdx.x;
    if (i < n) {
        float s = sp[0];
        float q = fminf(fmaxf(rintf(W[i] / s), -1.0f), 1.0f);
        Wq[i] = q * s;
    }
}

// ---------------------------------------------------------------------------
// Reductions for ternary scale: s = mean(|W|) + 1e-8
// ---------------------------------------------------------------------------
__global__ __launch_bounds__(256)
void k_absum_partial(const float* __restrict__ W, int n, float* __restrict__ part)
{
    __shared__ float sred[8];
    float s = 0.0f;
    for (int i = blockIdx.x * 256 + threadIdx.x; i < n; i += gridDim.x * 256)
        s += fabsf(W[i]);
    for (int off = 16; off > 0; off >>= 1) s += __shfl_xor(s, off, 32);
    if ((threadIdx.x & 31) == 0) sred[threadIdx.x >> 5] = s;
    __syncthreads();
    if (threadIdx.x == 0) {
        float t = 0.0f;
        for (int j = 0; j < 8; ++j) t += sred[j];
        part[blockIdx.x] = t;
    }
}

__global__ __launch_bounds__(256)
void k_absum_final(const float* __restrict__ part, int nb, float inv_n,
                   float* __restrict__ out)
{
    __shared__ float sred[8];
    float s = 0.0f;
    for (int i = threadIdx.x; i < nb; i += 256) s += part[i];
    for (int off = 16; off > 0; off >>= 1) s += __shfl_xor(s, off, 32);
    if ((threadIdx.x & 31) == 0) sred[threadIdx.x >> 5] = s;
    __syncthreads();
    if (threadIdx.x == 0) {
        float t = 0.0f;
        for (int j = 0; j < 8; ++j) t += sred[j];
        out[0] = t * inv_n + 1e-8f;
    }
}

// ---------------------------------------------------------------------------
// Selective SSM scan.  One thread per (b,d): N=16 hidden states in registers,
// sequential over S.  dA/dBx computed on the fly (saves 268 MB of HBM traffic
// vs. materializing like the reference).
// ---------------------------------------------------------------------------
__global__ __launch_bounds__(256)
void k_scan(const float* __restrict__ x, const float* __restrict__ delta,
            const float* __restrict__ Bm, const float* __restrict__ Cm,
            const float* __restrict__ A, const float* __restrict__ Dp,
            float* __restrict__ y)
{
    int t = blockIdx.x * blockDim.x + threadIdx.x;
    if (t >= BB * DDIM) return;
    int b = t / DDIM, d = t - b * DDIM;

    float h[NSTATE], Ad[NSTATE];
#pragma unroll
    for (int n = 0; n < NSTATE; ++n) { h[n] = 0.0f; Ad[n] = A[d * NSTATE + n]; }
    const float Dd = Dp[d];

    for (int s = 0; s < SS; ++s) {
        size_t idx = ((size_t)(b * SS + s)) * DDIM + d;
        float dt = delta[idx];
        float xv = x[idx];
        const float* __restrict__ Bs = Bm + (size_t)(b * SS + s) * NSTATE;
        const float* __restrict__ Cs = Cm + (size_t)(b * SS + s) * NSTATE;
        float yv = 0.0f;
#pragma unroll
        for (int n = 0; n < NSTATE; ++n) {
            float dA = expf(dt * Ad[n]);
            h[n] = dA * h[n] + dt * Bs[n] * xv;
            yv += Cs[n] * h[n];
        }
        y[idx] = yv + Dd * xv;
    }
}

// ---------------------------------------------------------------------------
// MHA: one wave per (b, head, query row), online softmax, head_dim=128
// => 4 floats per lane (wave32).
// ---------------------------------------------------------------------------
__global__ __launch_bounds__(256)
void k_attn(const float* __restrict__ q, const float* __restrict__ k,
            const float* __restrict__ v, const unsigned char* __restrict__ mask,
            float* __restrict__ o)
{
    int wid  = blockIdx.x * (blockDim.x >> 5) + (threadIdx.x >> 5);
    int lane = threadIdx.x & 31;
    if (wid >= BB * NHEADS * SS) return;
    int qi = wid % SS;
    int bh = wid / SS;
    int hh = bh % NHEADS;
    int b  = bh / NHEADS;

    const float scale = 0.08838834764831845f;  // 1/sqrt(128)
    size_t qoff = ((size_t)(b * SS + qi)) * DDIM + hh * HDIM + lane * 4;
    float4 qv = *(const float4*)(q + qoff);

    float mcur = -3.0e38f, l = 0.0f;
    float4 accv = {0.f, 0.f, 0.f, 0.f};

    for (int s = 0; s < SS; ++s) {
        size_t koff = ((size_t)(b * SS + s)) * DDIM + hh * HDIM + lane * 4;
        float4 kv = *(const float4*)(k + koff);
        float p = qv.x * kv.x + qv.y * kv.y + qv.z * kv.z + qv.w * kv.w;
        for (int off = 16; off > 0; off >>= 1) p += __shfl_xor(p, off, 32);
        float logit = p * scale;
        if (!mask[b * SS + s]) logit = -1e9f;
        float mnew = fmaxf(mcur, logit);
        float corr = expf(mcur - mnew);
        float w    = expf(logit - mnew);
        l = l * corr + w;
        float4 vv = *(const float4*)(v + koff);
        accv.x = accv.x * corr + w * vv.x;
        accv.y = accv.y * corr + w * vv.y;
        accv.z = accv.z * corr + w * vv.z;
        accv.w = accv.w * corr + w * vv.w;
        mcur = mnew;
    }
    float inv = 1.0f / l;
    float4 ov = {accv.x * inv, accv.y * inv, accv.z * inv, accv.w * inv};
    *(float4*)(o + qoff) = ov;
}

// ---------------------------------------------------------------------------
// Batched transpose (B, rows, cols) -> (B, cols, rows), 32x32 LDS tiles
// ---------------------------------------------------------------------------
__global__ __launch_bounds__(256)
void k_transpose(const float* __restrict__ in, float* __restrict__ out,
                 int rows, int cols)
{
    __shared__ float tile[32][33];
    int b = blockIdx.z;
    const float* ib = in  + (size_t)b * rows * cols;
    float*       ob = out + (size_t)b * rows * cols;
    int c0 = blockIdx.x * 32, r0 = blockIdx.y * 32;
    int tx = threadIdx.x & 31, ty = threadIdx.x >> 5;   // ty in [0,8)
#pragma unroll
    for (int j = 0; j < 4; ++j)
        tile[ty + j * 8][tx] = ib[(size_t)(r0 + ty + j * 8) * cols + c0 + tx];
    __syncthreads();
#pragma unroll
    for (int j = 0; j < 4; ++j)
        ob[(size_t)(c0 + ty + j * 8) * rows + r0 + tx] = tile[tx][ty + j * 8];
}

// ---------------------------------------------------------------------------
// Row kernels (one 256-thread block per (b,s) row of 512 elements)
// ---------------------------------------------------------------------------
__global__ __launch_bounds__(256)
void k_sparse_gate(float* __restrict__ delta)
{
    __shared__ float sred[8];
    int row = blockIdx.x, tid = threadIdx.x;
    size_t base = (size_t)row * DDIM;
    float v0 = delta[base + tid];
    float v1 = delta[base + tid + 256];
    float s = v0 + v1;
    for (int off = 16; off > 0; off >>= 1) s += __shfl_xor(s, off, 32);
    if ((tid & 31) == 0) sred[tid >> 5] = s;
    __syncthreads();
    float tot = 0.0f;
    for (int j = 0; j < 8; ++j) tot += sred[j];
    float mean = tot / (float)DDIM;
    delta[base + tid]       = (v0 > mean) ? v0 : 0.0f;
    delta[base + tid + 256] = (v1 > mean) ? v1 : 0.0f;
}

__global__ __launch_bounds__(256)
void k_act_quant(float* __restrict__ h)
{
    __shared__ float sred[8];
    int row = blockIdx.x, tid = threadIdx.x;
    size_t base = (size_t)row * DDIM;
    float v0 = h[base + tid];
    float v1 = h[base + tid + 256];
    float a = fmaxf(fabsf(v0), fabsf(v1));
    for (int off = 16; off > 0; off >>= 1) a = fmaxf(a, __shfl_xor(a, off, 32));
    if ((tid & 31) == 0) sred[tid >> 5] = a;
    __syncthreads();
    float mx = 0.0f;
    for (int j = 0; j < 8; ++j) mx = fmaxf(mx, sred[j]);
    float s = 127.0f / (mx + 1e-8f);
    float inv = 1.0f / s;
    h[base + tid]       = fminf(fmaxf(rintf(v0 * s), -128.0f), 127.0f) * inv;
    h[base + tid + 256] = fminf(fmaxf(rintf(v1 * s), -128.0f), 127.0f) * inv;
}

__global__ __launch_bounds__(256)
void k_final_ln(const float* __restrict__ fcout, const float* __restrict__ x,
                const unsigned char* __restrict__ mask,
                const float* __restrict__ scale, const float* __restrict__ bias,
                float* __restrict__ out)
{
    __shared__ float s1r[8], s2r[8];
    int row = blockIdx.x, tid = threadIdx.x;
    size_t base = (size_t)row * DDIM;
    float mk = mask[row] ? 1.0f : 0.0f;
    float v0 = fcout[base + tid]       + x[base + tid]       * mk;
    float v1 = fcout[base + tid + 256] + x[base + tid + 256] * mk;
    float s1 = v0 + v1, s2 = v0 * v0 + v1 * v1;
    for (int off = 16; off > 0; off >>= 1) {
        s1 += __shfl_xor(s1, off, 32);
        s2 += __shfl_xor(s2, off, 32);
    }
    if ((tid & 31) == 0) { s1r[tid >> 5] = s1; s2r[tid >> 5] = s2; }
    __syncthreads();
    float t1 = 0.0f, t2 = 0.0f;
    for (int j = 0; j < 8; ++j) { t1 += s1r[j]; t2 += s2r[j]; }
    float mu  = t1 / (float)DDIM;
    float var = t2 / (float)DDIM - mu * mu;
    float r = 1.0f / sqrtf(var + 1e-6f);
    out[base + tid]       = (v0 - mu) * r * scale[tid]       + bias[tid];
    out[base + tid + 256] = (v1 - mu) * r * scale[tid + 256] + bias[tid + 256];
}

// ---------------------------------------------------------------------------
// Host launcher
// ---------------------------------------------------------------------------
extern "C" void kernel_launch(void* const* d_in, const int* in_sizes, int n_in,
                              void* d_out, int out_size, void* d_ws, size_t ws_size,
                              hipStream_t stream)
{
    const float* x            = (const float*)d_in[0];
    const unsigned char* mask = (const unsigned char*)d_in[1];
    const float* m1_A   = (const float*)d_in[2];
    const float* m1_Wdt = (const float*)d_in[3];
    const float* m1_bdt = (const float*)d_in[4];
    const float* m1_WB  = (const float*)d_in[5];
    const float* m1_WC  = (const float*)d_in[6];
    const float* m1_D   = (const float*)d_in[7];
    const float* Wq     = (const float*)d_in[8];
    const float* Wk     = (const float*)d_in[9];
    const float* Wv     = (const float*)d_in[10];
    const float* Wo     = (const float*)d_in[11];
    const float* bo     = (const float*)d_in[12];
    const float* bn_W   = (const float*)d_in[13];
    const float* bn_b   = (const float*)d_in[14];
    const float* lq_W   = (const float*)d_in[15];
    const float* lq_b   = (const float*)d_in[16];
    const float* lq_tau = (const float*)d_in[17];
    const float* sm_A   = (const float*)d_in[18];
    const float* sm_Wdt = (const float*)d_in[19];
    const float* sm_bdt = (const float*)d_in[20];
    const float* sm_WB  = (const float*)d_in[21];
    const float* sm_WC  = (const float*)d_in[22];
    const float* sm_D   = (const float*)d_in[23];
    const float* m2_A   = (const float*)d_in[24];
    const float* m2_Wdt = (const float*)d_in[25];
    const float* m2_bdt = (const float*)d_in[26];
    const float* m2_WB  = (const float*)d_in[27];
    const float* m2_WC  = (const float*)d_in[28];
    const float* m2_D   = (const float*)d_in[29];
    const float* mb_Wg  = (const float*)d_in[30];
    const float* mb_Wm  = (const float*)d_in[31];
    const float* fc_W   = (const float*)d_in[32];
    const float* fc_b   = (const float*)d_in[33];
    const float* lnsc   = (const float*)d_in[34];
    const float* lnbs   = (const float*)d_in[35];
    float* out = (float*)d_out;

    float* W0   = (float*)d_ws;        // activation buffers, BSD floats each
    float* W1   = W0 + BSD;
    float* W2   = W1 + BSD;
    float* W3   = W2 + BSD;
    float* W4   = W3 + BSD;
    float* W5   = W4 + BSD;
    float* WQNT = W5 + BSD;            // SS*SS ternarized weights
    float* BMb  = WQNT + SS * SS;      // B*S*N
    float* CMb  = BMb + BSN;
    float* RED  = CMb + BSN;           // 1024 partials + scalar at RED+2048

    const int EW = (BSD + 255) / 256;
    const int ROWS = BB * SS;

    auto gemm = [&](const float* A, const float* Bw, const float* bias, float* C,
                    int M, int K, int N, int lda, int ldb, int ldc) {
        if ((M & 31) == 0 && (N & 31) == 0) {
            int tiles = (M >> 5) * (N >> 5);
            gemm_wmma<2, 2><<<(tiles + 7) / 8, 256, 0, stream>>>(
                A, Bw, bias, C, M, K, N, lda, ldb, ldc);
        } else {
            int tiles = (M >> 4) * (N >> 4);
            gemm_wmma<1, 1><<<(tiles + 7) / 8, 256, 0, stream>>>(
                A, Bw, bias, C, M, K, N, lda, ldb, ldc);
        }
    };
    auto scan = [&](const float* xin, float* delta, const float* Wdt, const float* bdt,
                    const float* WBp, const float* WCp, const float* Ap, const float* Dp,
                    float* yout, bool sparse) {
        gemm(xin, Wdt, bdt, delta, ROWS, DDIM, DDIM, DDIM, DDIM, DDIM);
        k_softplus<<<EW, 256, 0, stream>>>(delta, BSD);
        if (sparse) k_sparse_gate<<<ROWS, 256, 0, stream>>>(delta);
        gemm(xin, WBp, nullptr, BMb, ROWS, DDIM, NSTATE, DDIM, NSTATE, NSTATE);
        gemm(xin, WCp, nullptr, CMb, ROWS, DDIM, NSTATE, DDIM, NSTATE, NSTATE);
        k_scan<<<(BB * DDIM) / 256, 256, 0, stream>>>(xin, delta, BMb, CMb, Ap, Dp, yout);
    };

    // 1) xm = x * mask
    k_mask_mul<<<EW, 256, 0, stream>>>(x, mask, W0, BSD);

    // 2) mamba scan 1:  W0 -> W2
    scan(W0, W1, m1_Wdt, m1_bdt, m1_WB, m1_WC, m1_A, m1_D, W2, false);

    // 3) MHA:  W2 -> W0
    gemm(W2, Wq, nullptr, W3, ROWS, DDIM, DDIM, DDIM, DDIM, DDIM);
    gemm(W2, Wk, nullptr, W4, ROWS, DDIM, DDIM, DDIM, DDIM, DDIM);
    gemm(W2, Wv, nullptr, W5, ROWS, DDIM, DDIM, DDIM, DDIM, DDIM);
    k_attn<<<(BB * NHEADS * SS) / 8, 256, 0, stream>>>(W3, W4, W5, mask, W1);
    gemm(W1, Wo, bo, W0, ROWS, DDIM, DDIM, DDIM, DDIM, DDIM);

    // 4) BitNet-liquid stack (mean over 3 layers):  W0 -> W2
    k_transpose<<<dim3(DDIM / 32, SS / 32, BB), 256, 0, stream>>>(W0, W1, SS, DDIM); // hT (B,D,S)
    k_fill<<<EW, 256, 0, stream>>>(W2, 0.0f, BSD);
    for (int l = 0; l < NLAYERS; ++l) {
        const float* Wl = bn_W + (size_t)l * SS * SS;
        k_absum_partial<<<1024, 256, 0, stream>>>(Wl, SS * SS, RED);
        k_absum_final<<<1, 256, 0, stream>>>(RED, 1024, 1.0f / (float)(SS * SS), RED + 2048);
        k_ternary<<<(SS * SS + 255) / 256, 256, 0, stream>>>(Wl, RED + 2048, WQNT, SS * SS);
        // (B*D, S) @ (S, S) + b[t]  -> yt (B,D,S)
        gemm(W1, WQNT, bn_b + (size_t)l * SS, W3, BB * DDIM, SS, SS, SS, SS, SS);
        k_transpose<<<dim3(SS / 32, DDIM / 32, BB), 256, 0, stream>>>(W3, W4, DDIM, SS); // hbn (B,S,D)
        gemm(W4, lq_W + (size_t)l * DDIM * DDIM, lq_b + (size_t)l * DDIM, W5,
             ROWS, DDIM, DDIM, DDIM, DDIM, DDIM);
        k_liquid_acc<<<EW, 256, 0, stream>>>(W2, W4, W5, lq_tau + (size_t)l * DDIM, BSD);
    }
    k_scale<<<EW, 256, 0, stream>>>(W2, 1.0f / 3.0f, BSD);

    // 5) sparse mamba scan:  W2 -> W0
    scan(W2, W1, sm_Wdt, sm_bdt, sm_WB, sm_WC, sm_A, sm_D, W0, true);

    // 6) mamba scan 2:  W0 -> W2
    scan(W0, W1, m2_Wdt, m2_bdt, m2_WB, m2_WC, m2_A, m2_D, W2, false);

    // 7) act_quant (in place on W2)
    k_act_quant<<<ROWS, 256, 0, stream>>>(W2);

    // 8) MetaBAMDP gated update (in place on W2)
    gemm(W2, mb_Wg, nullptr, W0, ROWS, DDIM, DDIM, DDIM, DDIM, DDIM);
    gemm(W2, mb_Wm, nullptr, W1, ROWS, DDIM, DDIM, DDIM, DDIM, DDIM);
    k_metab<<<EW, 256, 0, stream>>>(W2, W0, W1, BSD);

    // 9) fc
    gemm(W2, fc_W, fc_b, W3, ROWS, DDIM, DDIM, DDIM, DDIM, DDIM);

    // 10) + residual (x*mask), LayerNorm -> out
    k_final_ln<<<ROWS, 256, 0, stream>>>(W3, x, mask, lnsc, lnbs, out);
}